// PtrNet1_2430951489702
// MI455X (gfx1250) — compile-verified
//
#include <hip/hip_runtime.h>
#include <math.h>

// ---------------- CDNA5 helpers ----------------
typedef __attribute__((ext_vector_type(2))) float v2f;
typedef __attribute__((ext_vector_type(8))) float v8f;

#define C_CLIPF 10.0f
#define NEGBIG  1e8f
#define Bsz 16
#define Nn  1000
#define Hh  128
#define BN  (Bsz*Nn)
#define PROJ_SZ ((size_t)BN*(size_t)Hh)   // floats per projection matrix

__device__ __forceinline__ float fast_tanh(float x){
#if __has_builtin(__builtin_amdgcn_tanhf)
    return __builtin_amdgcn_tanhf(x);        // v_tanh_f32 (TRANS op on CDNA5)
#else
    return tanhf(x);
#endif
}
__device__ __forceinline__ float fast_exp(float x){
#if __has_builtin(__builtin_amdgcn_exp2f)
    return __builtin_amdgcn_exp2f(x * 1.4426950408889634f);  // v_exp_f32
#else
    return __expf(x);
#endif
}
__device__ __forceinline__ float fast_log(float x){
#if __has_builtin(__builtin_amdgcn_logf)
    return __builtin_amdgcn_logf(x) * 0.6931471805599453f;   // v_log_f32
#else
    return logf(x);
#endif
}

// ---------------- Phase 1: projection GEMMs via WMMA f32 16x16x4 ----------------
// 13 independent [16000,128] x [128,128] GEMMs:
//   p = 0..11 : {Wref1,Vec1,Wref3,Vec3,Wref4,Vec4} x heads {0,1}
//   p = 12    : Wref2
struct GemmArgs {
    const float* A;        // enc_h, [16000,128] row-major
    const float* W[13];    // each [128,128] row-major (d-major, h cols)
    float*       Out;      // [13][16000][128]
};

__global__ __launch_bounds__(256) void proj_gemm(GemmArgs ga)
{
    // A tile in LDS, stride 129 to avoid bank conflicts on column reads
    __shared__ float At[16 * 129];
    const int rt  = blockIdx.x;   // row tile (16 rows)
    const int p   = blockIdx.y;   // projection id
    const int tid = threadIdx.x;

    const float* A = ga.A + (size_t)rt * 16 * Hh;
    for (int i = tid; i < 512; i += 256) {         // 512 float4 = 16x128 tile
        const int r = i >> 5;
        const int c = (i & 31) << 2;
        const float4 f = *(const float4*)(A + (size_t)r * Hh + c);
        At[r * 129 + c + 0] = f.x;
        At[r * 129 + c + 1] = f.y;
        At[r * 129 + c + 2] = f.z;
        At[r * 129 + c + 3] = f.w;
    }
    __syncthreads();

    const int wave = tid >> 5;          // 8 waves -> 8 column tiles of 16
    const int lane = tid & 31;
    const int colT = wave;
    const int mrow = lane & 15;                 // A-matrix: M = lane%16
    const int koff = (lane < 16) ? 0 : 2;       // lanes 16-31 hold K+2,K+3
    const int ncol = lane & 15;                 // B/D: N = lane%16

    const float* W = ga.W[p];
    v8f acc = {0.f,0.f,0.f,0.f,0.f,0.f,0.f,0.f};

    for (int k = 0; k < Hh; k += 4) {
        v2f a, bm;
        a.x  = At[mrow * 129 + k + koff];
        a.y  = At[mrow * 129 + k + koff + 1];
        bm.x = W[(size_t)(k + koff)     * Hh + colT * 16 + ncol];
        bm.y = W[(size_t)(k + koff + 1) * Hh + colT * 16 + ncol];
        acc = __builtin_amdgcn_wmma_f32_16x16x4_f32(
                  false, a, false, bm, (short)0, acc, false, false);
    }

    // D layout: VGPR r -> M = r + (lane<16 ? 0 : 8), N = lane%16
    float* O = ga.Out + (size_t)p * PROJ_SZ + (size_t)rt * 16 * Hh + colT * 16 + ncol;
    const int mbase = (lane < 16) ? 0 : 8;
#pragma unroll
    for (int r = 0; r < 8; ++r)
        O[(size_t)(r + mbase) * Hh] = acc[r];
}

// ---------------- Phase 2: fused persistent decoder ----------------
// One workgroup per batch element b (fully independent chains).
// 1024 threads = 32 waves. Per glimpse: 16 waves per head, each wave owns
// strided rows n; single-pass online softmax fused with g = sum a*r.
__global__ __launch_bounds__(1024) void decode_kernel(
    const float* __restrict__ enc_h, const float* __restrict__ mask0,
    const float* __restrict__ Wq1,  const float* __restrict__ Wq3,
    const float* __restrict__ Wq4,  const float* __restrict__ Wmh,
    const float* __restrict__ Wq2,  const float* __restrict__ Vec2,
    const float* __restrict__ dec_inp, const int* __restrict__ nd_ptr,
    const float* __restrict__ proj, float* __restrict__ out)
{
    const int b    = blockIdx.x;
    const int tid  = threadIdx.x;
    const int wave = tid >> 5;
    const int lane = tid & 31;

    __shared__ float maskS[Nn];                 // 4 KB
    __shared__ float uS[Nn];                    // 4 KB
    __shared__ float hmean[Hh];
    __shared__ float prev[Hh];
    __shared__ float q0[2 * Hh];
    __shared__ float qm[2][Hh];
    __shared__ float gg[2][Hh];
    __shared__ float qn[Hh];
    __shared__ __align__(16) float qw[Hh];
    __shared__ float mxS[32];
    __shared__ float smS[32];
    __shared__ int   idxS[32];
    __shared__ float gpart[32][Hh];             // 16 KB
    __shared__ int   selS;

    const float* encB = enc_h + (size_t)b * Nn * Hh;

    // ---- h_mean over N ----
    {
        const int h    = tid & (Hh - 1);
        const int part = tid >> 7;              // 0..7
        float s = 0.f;
        for (int n = part; n < Nn; n += 8) s += encB[(size_t)n * Hh + h];
        gpart[part][h] = s;
        __syncthreads();
        if (tid < Hh) {
            float t = 0.f;
            for (int pp = 0; pp < 8; ++pp) t += gpart[pp][tid];
            hmean[tid] = t * (1.0f / (float)Nn);
        }
    }
    for (int i = tid; i < Nn; i += 1024) maskS[i] = mask0[(size_t)b * Nn + i];
    if (tid < Hh) prev[tid] = dec_inp[tid];
    __syncthreads();
    const int nd = *nd_ptr;

    for (int step = 0; step < nd; ++step) {
        if (tid < Hh) { q0[tid] = hmean[tid]; q0[Hh + tid] = prev[tid]; }
        __syncthreads();

        for (int gi = 0; gi < 3; ++gi) {
            // ---- query projection: qm[m][h] = sum_i qin[i] * Wq[m,i,h] ----
            const int    qlen = (gi == 0) ? 2 * Hh : Hh;
            const float* qin  = (gi == 0) ? q0 : qn;
            const float* Wqp  = (gi == 0) ? Wq1 : ((gi == 1) ? Wq3 : Wq4);
            if (tid < 2 * Hh) {
                const int m = tid >> 7, h = tid & (Hh - 1);
                const float* wcol = Wqp + (size_t)m * qlen * Hh + h;
                float acc = 0.f;
                for (int i = 0; i < qlen; ++i) acc += qin[i] * wcol[(size_t)i * Hh];
                qm[m][h] = acc;
            }
            __syncthreads();

            // ---- glimpse core: online softmax + fused g accumulation ----
            const int m  = wave & 1;            // head
            const int h0 = lane << 2;           // 4 h-values per lane
            const float* rb = proj + (size_t)(gi * 4 + m)     * PROJ_SZ + (size_t)b * Nn * Hh;
            const float* vb = proj + (size_t)(gi * 4 + 2 + m) * PROJ_SZ + (size_t)b * Nn * Hh;
            const float q0v = qm[m][h0], q1v = qm[m][h0 + 1];
            const float q2v = qm[m][h0 + 2], q3v = qm[m][h0 + 3];
            float mx = -3.0e38f, sm = 0.f;
            float g0 = 0.f, g1 = 0.f, g2 = 0.f, g3 = 0.f;

            for (int n = (wave >> 1); n < Nn; n += 16) {
                if (maskS[n] > 0.5f) continue;  // masked -> softmax weight 0
                const float4 r = *(const float4*)(rb + (size_t)n * Hh + h0);
                const float4 v = *(const float4*)(vb + (size_t)n * Hh + h0);
                float p = fast_tanh(q0v + r.x) * v.x;
                p      += fast_tanh(q1v + r.y) * v.y;
                p      += fast_tanh(q2v + r.z) * v.z;
                p      += fast_tanh(q3v + r.w) * v.w;
#pragma unroll
                for (int off = 16; off > 0; off >>= 1) p += __shfl_xor(p, off, 32);
                const float s  = C_CLIPF * fast_tanh(p);
                const float nm = fmaxf(mx, s);
                const float c  = fast_exp(mx - nm);
                const float w  = fast_exp(s - nm);
                sm = sm * c + w;
                g0 = g0 * c + w * r.x;  g1 = g1 * c + w * r.y;
                g2 = g2 * c + w * r.z;  g3 = g3 * c + w * r.w;
                mx = nm;
            }
            if (lane == 0) { mxS[wave] = mx; smS[wave] = sm; }
            gpart[wave][h0] = g0; gpart[wave][h0 + 1] = g1;
            gpart[wave][h0 + 2] = g2; gpart[wave][h0 + 3] = g3;
            __syncthreads();

            // ---- merge 16 wave-partials per head ----
            if (tid < 2 * Hh) {
                const int m2 = tid >> 7, h = tid & (Hh - 1);
                float MX = -3.0e38f;
                for (int w2 = m2; w2 < 32; w2 += 2) MX = fmaxf(MX, mxS[w2]);
                float SUM = 0.f, G = 0.f;
                for (int w2 = m2; w2 < 32; w2 += 2) {
                    const float cc = fast_exp(mxS[w2] - MX);
                    SUM += smS[w2] * cc;
                    G   += gpart[w2][h] * cc;
                }
                gg[m2][h] = G / SUM;
            }
            __syncthreads();

            // ---- merge heads: qn = concat(gg0, gg1) @ Wmh ----
            if (tid < Hh) {
                float acc = 0.f;
                for (int j = 0; j < 2 * Hh; ++j)
                    acc += gg[j >> 7][j & (Hh - 1)] * Wmh[(size_t)j * Hh + tid];
                qn[tid] = acc;
            }
            __syncthreads();
        }

        // ---- pointer logits: u[n] = Vec2 . tanh(qn@Wq2 + r2[n]) ----
        if (tid < Hh) {
            float acc = 0.f;
            for (int i = 0; i < Hh; ++i) acc += qn[i] * Wq2[(size_t)i * Hh + tid];
            qw[tid] = acc;
        }
        __syncthreads();
        {
            const int h0 = lane << 2;
            const float4 ve = *(const float4*)(Vec2 + h0);
            const float4 qv = *(const float4*)(qw + h0);
            const float* r2b = proj + (size_t)12 * PROJ_SZ + (size_t)b * Nn * Hh;
            for (int n = wave; n < Nn; n += 32) {
                const float4 r = *(const float4*)(r2b + (size_t)n * Hh + h0);
                float p = fast_tanh(qv.x + r.x) * ve.x + fast_tanh(qv.y + r.y) * ve.y
                        + fast_tanh(qv.z + r.z) * ve.z + fast_tanh(qv.w + r.w) * ve.w;
#pragma unroll
                for (int off = 16; off > 0; off >>= 1) p += __shfl_xor(p, off, 32);
                if (lane == 0)
                    uS[n] = C_CLIPF * fast_tanh(p) - NEGBIG * maskS[n];
            }
        }
        __syncthreads();

        // ---- argmax (first-max semantics) ----
        float lmx = -3.0e38f; int li = 0;
        for (int n = tid; n < Nn; n += 1024) {
            const float u = uS[n];
            if (u > lmx) { lmx = u; li = n; }
        }
#pragma unroll
        for (int off = 16; off > 0; off >>= 1) {
            const float om = __shfl_xor(lmx, off, 32);
            const int   oi = __shfl_xor(li, off, 32);
            if (om > lmx || (om == lmx && oi < li)) { lmx = om; li = oi; }
        }
        if (lane == 0) { mxS[wave] = lmx; idxS[wave] = li; }
        __syncthreads();
        float MX = -3.0e38f; int SI = 0;
        for (int w2 = 0; w2 < 32; ++w2) {
            const float vm = mxS[w2]; const int vi = idxS[w2];
            if (vm > MX || (vm == MX && vi < SI)) { MX = vm; SI = vi; }
        }

        // ---- log-softmax ----
        float ps = 0.f;
        for (int n = tid; n < Nn; n += 1024) ps += fast_exp(uS[n] - MX);
#pragma unroll
        for (int off = 16; off > 0; off >>= 1) ps += __shfl_xor(ps, off, 32);
        if (lane == 0) smS[wave] = ps;
        __syncthreads();
        float SUM = 0.f;
        for (int w2 = 0; w2 < 32; ++w2) SUM += smS[w2];
        const float LSE = MX + fast_log(SUM);

        float* lp = out + ((size_t)step * Bsz + b) * Nn;
        for (int n = tid; n < Nn; n += 1024) lp[n] = uS[n] - LSE;

        if (tid == 0) {
            selS = SI;
            maskS[SI] = 1.0f;
            out[(size_t)nd * Bsz * Nn + (size_t)step * Bsz + b] = (float)SI;
        }
        __syncthreads();
        if (tid < Hh) prev[tid] = encB[(size_t)selS * Hh + tid];
        __syncthreads();
    }
}

// ---------------- host launch ----------------
extern "C" void kernel_launch(void* const* d_in, const int* in_sizes, int n_in,
                              void* d_out, int out_size, void* d_ws, size_t ws_size,
                              hipStream_t stream)
{
    (void)in_sizes; (void)n_in; (void)out_size; (void)ws_size;
    const float* enc_h   = (const float*)d_in[0];
    const float* mask0   = (const float*)d_in[1];
    const float* Wq1     = (const float*)d_in[2];
    const float* Wref1   = (const float*)d_in[3];
    const float* Vec1    = (const float*)d_in[4];
    const float* Wq3     = (const float*)d_in[5];
    const float* Wref3   = (const float*)d_in[6];
    const float* Vec3    = (const float*)d_in[7];
    const float* Wq4     = (const float*)d_in[8];
    const float* Wref4   = (const float*)d_in[9];
    const float* Vec4    = (const float*)d_in[10];
    const float* Wmh     = (const float*)d_in[11];
    const float* Wq2     = (const float*)d_in[12];
    const float* Wref2   = (const float*)d_in[13];
    const float* Vec2    = (const float*)d_in[14];
    const float* dec_inp = (const float*)d_in[15];
    const int*   nd      = (const int*)d_in[16];

    // workspace: 13 projection matrices of [16000,128] f32 (~107 MB, L2-resident)
    float* proj = (float*)d_ws;

    GemmArgs ga;
    ga.A = enc_h;
    const int HH = Hh * Hh;
    ga.W[0]  = Wref1;  ga.W[1]  = Wref1 + HH;
    ga.W[2]  = Vec1;   ga.W[3]  = Vec1  + HH;
    ga.W[4]  = Wref3;  ga.W[5]  = Wref3 + HH;
    ga.W[6]  = Vec3;   ga.W[7]  = Vec3  + HH;
    ga.W[8]  = Wref4;  ga.W[9]  = Wref4 + HH;
    ga.W[10] = Vec4;   ga.W[11] = Vec4  + HH;
    ga.W[12] = Wref2;
    ga.Out = proj;

    proj_gemm<<<dim3(BN / 16, 13), 256, 0, stream>>>(ga);
    decode_kernel<<<Bsz, 1024, 0, stream>>>(enc_h, mask0, Wq1, Wq3, Wq4,
                                            Wmh, Wq2, Vec2, dec_inp, nd,
                                            proj, (float*)d_out);
}